// VectorQuantizer_27513560498742
// MI455X (gfx1250) — compile-verified
//
#include <hip/hip_runtime.h>
#include <math.h>

typedef __attribute__((ext_vector_type(2))) float v2f;
typedef __attribute__((ext_vector_type(8))) float v8f;

#define C_DIM        64
#define K_CODES      512
#define HW           4096
#define N_TOTAL      131072
#define ROWS_PER_BLK 128
#define LDS_STRIDE   65
#define PACK_OFS     1536        // floats; 16B-aligned. pack = 32768 floats (128KB)

// ws layout (floats): [0..511] ||e_k||^2, [512..1023] counts, [1024] loss accum,
//                     [1536..34303] WMMA-lane-packed embed: pack[j][k2][lane][4]

__global__ __launch_bounds__(256) void vq_prep(const float* __restrict__ embed,
                                               float* __restrict__ ws) {
    int gid = blockIdx.x * 256 + threadIdx.x;      // grid = 128 blocks -> 32768
    if (gid < K_CODES) {
        float s = 0.0f;
        #pragma unroll
        for (int c = 0; c < C_DIM; ++c) {
            float e = embed[c * K_CODES + gid];
            s += e * e;
        }
        ws[gid] = s;
        ws[K_CODES + gid] = 0.0f;                  // zero counts
    }
    if (gid == 0) ws[2 * K_CODES] = 0.0f;          // zero loss accumulator

    // Repack embed into per-lane WMMA order so the hot loop uses b128 loads:
    // element q of lane at (j, k2): wmma step k = 2*k2 + (q>>1),
    // B row = 4*k + 2*half + (q&1), B col = 16*j + lrow.
    int q    = gid & 3;
    int lane = (gid >> 2) & 31;
    int k2   = (gid >> 7) & 7;
    int j    = gid >> 10;
    int half = lane >> 4;
    int lrow = lane & 15;
    int k    = 2 * k2 + (q >> 1);
    int row  = 4 * k + 2 * half + (q & 1);
    int col  = 16 * j + lrow;
    ws[PACK_OFS + gid] = embed[row * K_CODES + col];
}

__global__ __launch_bounds__(256) void vq_main(const float* __restrict__ x,
                                               const float* __restrict__ embed,
                                               float* __restrict__ ws,
                                               float* __restrict__ out) {
    __shared__ float xlds[ROWS_PER_BLK * LDS_STRIDE];   // [nl][c], padded
    __shared__ int   idxlds[ROWS_PER_BLK];
    __shared__ float redlds[8];

    const int t   = threadIdx.x;
    const int n0  = blockIdx.x * ROWS_PER_BLK;     // first flat row of this block
    const int b   = n0 >> 12;                      // n0 / 4096
    const int hw0 = n0 & (HW - 1);
    const float* xb = x + (size_t)b * C_DIM * HW + hw0;

    // Stage x tile into LDS (transpose [c][nl] -> xlds[nl*S + c]); coalesced float4 reads.
    #pragma unroll
    for (int i = 0; i < 8; ++i) {
        int f4 = t + i * 256;                      // 2048 float4 = 8192 floats
        int e0 = f4 * 4;
        int c  = e0 >> 7;                          // /128
        int nl = e0 & 127;
        const float4 v = *reinterpret_cast<const float4*>(xb + (size_t)c * HW + nl);
        xlds[(nl + 0) * LDS_STRIDE + c] = v.x;
        xlds[(nl + 1) * LDS_STRIDE + c] = v.y;
        xlds[(nl + 2) * LDS_STRIDE + c] = v.z;
        xlds[(nl + 3) * LDS_STRIDE + c] = v.w;
    }
    __syncthreads();

    const int lane    = t & 31;
    const int wave    = t >> 5;
    const int half    = lane >> 4;                 // 0 or 1
    const int lrow    = lane & 15;
    const int rowBase = wave * 16;                 // 16 rows per wave
    const int kofs    = half * 2;                  // K offset within each 4-chunk

    // A-matrix (16x64 fp32) in registers: v2f per 4-wide K chunk.
    // Lane L<16: {a[M=L][4k+0], a[M=L][4k+1]}; lane L>=16: {a[M][4k+2], a[M][4k+3]}.
    v2f a[16];
    const float* arow = &xlds[(rowBase + lrow) * LDS_STRIDE];
    #pragma unroll
    for (int k = 0; k < 16; ++k) {
        v2f av;
        av.x = arow[4 * k + kofs];
        av.y = arow[4 * k + kofs + 1];
        a[k] = av;
    }

    float best[8];
    int   bidx[8];
    #pragma unroll
    for (int r = 0; r < 8; ++r) { best[r] = 3.402823e38f; bidx[r] = 0; }

    // Sweep 32 code tiles; per tile: 8 coalesced b128 loads feed 16 wmma 16x16x4.
    const float* packLane = ws + PACK_OFS + lane * 4;
    for (int j = 0; j < K_CODES / 16; ++j) {
        v8f acc = {};
        const float* bp = packLane + j * 1024;     // 1024 floats per code tile
        #pragma unroll
        for (int k2 = 0; k2 < 8; ++k2) {
            const float4 bv4 = *reinterpret_cast<const float4*>(bp + k2 * 128);
            v2f b0; b0.x = bv4.x; b0.y = bv4.y;
            v2f b1; b1.x = bv4.z; b1.y = bv4.w;
            acc = __builtin_amdgcn_wmma_f32_16x16x4_f32(
                false, a[2 * k2 + 0], false, b0, (short)0, acc, false, false);
            acc = __builtin_amdgcn_wmma_f32_16x16x4_f32(
                false, a[2 * k2 + 1], false, b1, (short)0, acc, false, false);
        }
        const int   code = j * 16 + lrow;
        const float en   = ws[code];
        #pragma unroll
        for (int r = 0; r < 8; ++r) {
            float d = en - 2.0f * acc[r];          // ||e||^2 - 2 x.e  (||x||^2 const)
            if (d < best[r]) { best[r] = d; bidx[r] = code; }
        }
    }

    // Min-reduce (value, index) across the 16-lane group; tie -> smaller index (argmin-first).
    #pragma unroll
    for (int m = 1; m <= 8; m <<= 1) {
        #pragma unroll
        for (int r = 0; r < 8; ++r) {
            float ov = __shfl_xor(best[r], m, 32);
            int   oi = __shfl_xor(bidx[r], m, 32);
            if (ov < best[r] || (ov == best[r] && oi < bidx[r])) {
                best[r] = ov; bidx[r] = oi;
            }
        }
    }

    float* counts = ws + K_CODES;
    if (lrow == 0) {                               // lanes 0 and 16: rows r / 8+r
        #pragma unroll
        for (int r = 0; r < 8; ++r) {
            int row = rowBase + half * 8 + r;
            idxlds[row] = bidx[r];
            out[(size_t)1 + (size_t)N_TOTAL * C_DIM + 1 + (n0 + row)] = (float)bidx[r];
            atomicAdd(&counts[bidx[r]], 1.0f);
        }
    }
    __syncthreads();

    // Phase 2: write quantized_st = x + (q - x) to [B,C,H,W]; accumulate (q-x)^2.
    float lsum = 0.0f;
    float* qout = out + 1;
    #pragma unroll
    for (int i = 0; i < 32; ++i) {
        int f  = t + i * 256;                      // 8192 elements
        int c  = f >> 7;
        int nl = f & 127;
        float xv  = xlds[nl * LDS_STRIDE + c];
        float q   = embed[c * K_CODES + idxlds[nl]];
        float dlt = q - xv;
        qout[(size_t)b * C_DIM * HW + (size_t)c * HW + hw0 + nl] = xv + dlt;
        lsum += dlt * dlt;
    }
    #pragma unroll
    for (int m = 1; m < 32; m <<= 1) lsum += __shfl_xor(lsum, m, 32);
    if (lane == 0) redlds[wave] = lsum;
    __syncthreads();
    if (t == 0) {
        float s = 0.0f;
        #pragma unroll
        for (int w = 0; w < 8; ++w) s += redlds[w];
        atomicAdd(&ws[2 * K_CODES], s);
    }
}

__global__ __launch_bounds__(512) void vq_final(const float* __restrict__ ws,
                                                float* __restrict__ out) {
    __shared__ float red[512];
    int k = threadIdx.x;
    float avg = ws[K_CODES + k] * (1.0f / (float)N_TOTAL);
    red[k] = avg * logf(avg + 1e-10f);
    __syncthreads();
    for (int s = 256; s > 0; s >>= 1) {
        if (k < s) red[k] += red[k + s];
        __syncthreads();
    }
    if (k == 0) {
        // loss = q_latent + 0.25*e_latent = 1.25 * mean((q - x)^2)
        out[0] = 1.25f * ws[2 * K_CODES] / (float)((size_t)N_TOTAL * C_DIM);
        out[(size_t)1 + (size_t)N_TOTAL * C_DIM] = expf(-red[0]);
    }
}

extern "C" void kernel_launch(void* const* d_in, const int* in_sizes, int n_in,
                              void* d_out, int out_size, void* d_ws, size_t ws_size,
                              hipStream_t stream) {
    const float* x     = (const float*)d_in[0];   // [32, 64, 64, 64]
    const float* embed = (const float*)d_in[1];   // [64, 512]
    float* out = (float*)d_out;                   // loss | quantized | perplexity | indices
    float* ws  = (float*)d_ws;

    vq_prep<<<128, 256, 0, stream>>>(embed, ws);
    vq_main<<<N_TOTAL / ROWS_PER_BLK, 256, 0, stream>>>(x, embed, ws, out);
    vq_final<<<1, 512, 0, stream>>>(ws, out);
}